// sknn_cv_37675453120700
// MI455X (gfx1250) — compile-verified
//
#include <hip/hip_runtime.h>
#include <math.h>

#define N      8192
#define D      5
#define TILE   16
#define NT     (N / TILE)      // 512 tiles per dimension
#define SPLIT  8               // j-dimension split per row-tile
#define WPB    8               // waves per block
#define BLOCK  (WPB * 32)

typedef __attribute__((ext_vector_type(2))) float v2f;
typedef __attribute__((ext_vector_type(8))) float v8f;

// ---------------------------------------------------------------------------
// Kernel 1: per-row squared norms, zero accumulators and output.
// ---------------------------------------------------------------------------
__global__ void sknn_init_kernel(const float* __restrict__ Radio,
                                 float* __restrict__ sq,
                                 float* __restrict__ sumw,
                                 float* __restrict__ sumx,
                                 float* __restrict__ sumy,
                                 float* __restrict__ out) {
    int i = blockIdx.x * blockDim.x + threadIdx.x;
    if (i < N) {
        float s = 0.f;
#pragma unroll
        for (int k = 0; k < D; ++k) {
            float r = Radio[i * D + k];
            s = fmaf(r, r, s);
        }
        sq[i]   = s;
        sumw[i] = 0.f;
        sumx[i] = 0.f;
        sumy[i] = 0.f;
    }
    if (i == 0) out[0] = 0.f;
}

// ---------------------------------------------------------------------------
// Kernel 2: Gram tiles via V_WMMA_F32_16X16X4_F32, fused exp + weighted
// accumulation. One wave = one (16-row tile, j-split) unit.
//
// f32 16x4 A-operand layout == 4x16 B-operand layout for a Gram matrix:
//   lane L holds row base+(L%16); VGPR0 = K(L<16?0:2), VGPR1 = K(L<16?1:3).
// C layout: lane L, VGPR v -> (m = v + 8*(L>=16), n = L%16).
// ---------------------------------------------------------------------------
__global__ void __launch_bounds__(BLOCK)
sknn_gram_kernel(const float* __restrict__ Radio,
                 const float* __restrict__ Loc,
                 const float* __restrict__ sigma,
                 const float* __restrict__ sq,
                 float* __restrict__ sumw,
                 float* __restrict__ sumx,
                 float* __restrict__ sumy) {
    const int lane   = threadIdx.x & 31;
    const int wib    = threadIdx.x >> 5;
    const int unit   = blockIdx.x * WPB + wib;
    const int rtile  = unit / SPLIT;
    const int jsplit = unit % SPLIT;
    const int i0     = rtile * TILE;
    const int half   = lane >> 4;   // 0: lanes 0-15, 1: lanes 16-31
    const int ln     = lane & 15;

    const float s      = sigma[0];
    const float inv2s2 = 1.0f / (2.0f * s * s);

    // A operands for this wave's 16 rows (fixed across the j loop).
    const int arow = i0 + ln;
    v2f a, a2;
    a.x  = Radio[arow * D + (half ? 2 : 0)];
    a.y  = Radio[arow * D + (half ? 3 : 1)];
    a2.x = half ? 0.f : Radio[arow * D + 4];   // K=4 (K=5..7 zero-padded)
    a2.y = 0.f;

    // Squared norms of the 8 rows this lane's C-accumulators cover.
    float sqi[8];
#pragma unroll
    for (int v = 0; v < 8; ++v) sqi[v] = sq[i0 + half * 8 + v];

    float accw[8], accx[8], accy[8];
#pragma unroll
    for (int v = 0; v < 8; ++v) { accw[v] = 0.f; accx[v] = 0.f; accy[v] = 0.f; }

    const int jt0 = jsplit * (NT / SPLIT);
    for (int t = 0; t < NT / SPLIT; ++t) {
        const int j0   = (jt0 + t) * TILE;
        const int brow = j0 + ln;      // this lane's global column index

        v2f b, b2;
        b.x  = Radio[brow * D + (half ? 2 : 0)];
        b.y  = Radio[brow * D + (half ? 3 : 1)];
        b2.x = half ? 0.f : Radio[brow * D + 4];
        b2.y = 0.f;

        const float sqj = sq[brow];
        const float lx  = Loc[brow * 2 + 0];
        const float ly  = Loc[brow * 2 + 1];

        v8f c = {};
        c = __builtin_amdgcn_wmma_f32_16x16x4_f32(false, a,  false, b,
                                                  (short)0, c, false, false);
        c = __builtin_amdgcn_wmma_f32_16x16x4_f32(false, a2, false, b2,
                                                  (short)0, c, false, false);

#pragma unroll
        for (int v = 0; v < 8; ++v) {
            float d2 = fmaf(-2.f, c[v], sqi[v] + sqj);
            d2 = fmaxf(d2, 0.f);
            float w = __expf(-d2 * inv2s2);
            const int gi = i0 + half * 8 + v;
            w = (gi == brow) ? 0.f : w;     // leave-one-out diagonal mask
            accw[v] += w;
            accx[v] = fmaf(w, lx, accx[v]);
            accy[v] = fmaf(w, ly, accy[v]);
        }
    }

    // Reduce over the 16 column-lanes that share each row (masks < 16 stay
    // inside each 16-lane half of the wave32).
#pragma unroll
    for (int v = 0; v < 8; ++v) {
#pragma unroll
        for (int m = 8; m >= 1; m >>= 1) {
            accw[v] += __shfl_xor(accw[v], m, 32);
            accx[v] += __shfl_xor(accx[v], m, 32);
            accy[v] += __shfl_xor(accy[v], m, 32);
        }
    }
    if (ln == 0) {
#pragma unroll
        for (int v = 0; v < 8; ++v) {
            const int gi = i0 + half * 8 + v;
            atomicAdd(&sumw[gi], accw[v]);
            atomicAdd(&sumx[gi], accx[v]);
            atomicAdd(&sumy[gi], accy[v]);
        }
    }
}

// ---------------------------------------------------------------------------
// Kernel 3: per-row prediction error, mean via block reduce + atomics.
// ---------------------------------------------------------------------------
__global__ void sknn_finalize_kernel(const float* __restrict__ Loc,
                                     const float* __restrict__ sumw,
                                     const float* __restrict__ sumx,
                                     const float* __restrict__ sumy,
                                     float* __restrict__ out) {
    __shared__ float red[256];
    int i = blockIdx.x * blockDim.x + threadIdx.x;
    float e = 0.f;
    if (i < N) {
        float invw = 1.0f / sumw[i];
        float dx = fmaf(sumx[i], invw, -Loc[i * 2 + 0]);
        float dy = fmaf(sumy[i], invw, -Loc[i * 2 + 1]);
        e = sqrtf(dx * dx + dy * dy);
    }
    red[threadIdx.x] = e;
    __syncthreads();
    for (int st = 128; st > 0; st >>= 1) {
        if ((int)threadIdx.x < st) red[threadIdx.x] += red[threadIdx.x + st];
        __syncthreads();
    }
    if (threadIdx.x == 0) atomicAdd(out, red[0] * (1.0f / (float)N));
}

// ---------------------------------------------------------------------------
extern "C" void kernel_launch(void* const* d_in, const int* in_sizes, int n_in,
                              void* d_out, int out_size, void* d_ws, size_t ws_size,
                              hipStream_t stream) {
    const float* Radio = (const float*)d_in[0];
    const float* Loc   = (const float*)d_in[1];
    const float* sigma = (const float*)d_in[2];
    float* out = (float*)d_out;

    float* ws   = (float*)d_ws;   // needs 4*N floats = 128 KB
    float* sq   = ws;
    float* sumw = ws + 1 * N;
    float* sumx = ws + 2 * N;
    float* sumy = ws + 3 * N;

    sknn_init_kernel<<<(N + 255) / 256, 256, 0, stream>>>(Radio, sq, sumw, sumx,
                                                          sumy, out);

    const int units  = NT * SPLIT;            // 4096 wave-units
    const int blocks = units / WPB;           // 512 blocks of 8 waves
    sknn_gram_kernel<<<blocks, BLOCK, 0, stream>>>(Radio, Loc, sigma, sq,
                                                   sumw, sumx, sumy);

    sknn_finalize_kernel<<<(N + 255) / 256, 256, 0, stream>>>(Loc, sumw, sumx,
                                                              sumy, out);
}